// AtlasJodie_31911607009496
// MI455X (gfx1250) — compile-verified
//
#include <hip/hip_runtime.h>

// Problem constants (match reference)
#define NN  1000000
#define BB  8192
#define DD  128
#define DEE 128
#define DTT 64
#define K1  448   // 256 mail + 64 time-feat + 128 prev_mem
#define K2  256   // [src | dst] stacked for the predictor
#define LN_EPS 1e-5f

typedef float v2f __attribute__((ext_vector_type(2)));
typedef float v8f __attribute__((ext_vector_type(8)));

// D(16x16,f32) = A(16x4,f32) * B(4x16,f32) + C   -- exact f32 matrix op (CDNA5)
__device__ __forceinline__ v8f wmma4(v2f a, v2f b, v8f c) {
  return __builtin_amdgcn_wmma_f32_16x16x4_f32(false, a, false, b, (short)0, c,
                                               false, false);
}

// ---------------------------------------------------------------------------
// Weight prep: transposed AND k-paired so each WMMA B-fragment is a single
// coalesced global_load_b64:  Wt2[p][n] = { W[2p][n], W[2p+1][n] }.
// ---------------------------------------------------------------------------
__global__ void prep_weights(const float* __restrict__ W_ih,
                             const float* __restrict__ W_hh,
                             const float* __restrict__ src_w,
                             const float* __restrict__ dst_w,
                             float2* __restrict__ Wt2,
                             float2* __restrict__ Wp2) {
  int idx = blockIdx.x * blockDim.x + threadIdx.x;
  if (idx < (K1 / 2) * DD) {
    int p = idx / DD, n = idx % DD;
    int k0 = p * 2, k1 = k0 + 1;   // 320 is even -> k0,k1 on same side of split
    float a = (k0 < 320) ? W_ih[n * 320 + k0] : W_hh[n * DD + (k0 - 320)];
    float b = (k1 < 320) ? W_ih[n * 320 + k1] : W_hh[n * DD + (k1 - 320)];
    Wt2[idx] = make_float2(a, b);
  } else {
    int j = idx - (K1 / 2) * DD;
    if (j < (K2 / 2) * DD) {
      int p = j / DD, n = j % DD;
      int k0 = p * 2, k1 = k0 + 1;
      float a = (k0 < DD) ? src_w[n * DD + k0] : dst_w[n * DD + (k0 - DD)];
      float b = (k1 < DD) ? src_w[n * DD + k1] : dst_w[n * DD + (k1 - DD)];
      Wp2[j] = make_float2(a, b);
    }
  }
}

// ---------------------------------------------------------------------------
// Fused: gather -> time feature -> WMMA GEMM (16x448 @ 448x128) -> tanh ->
// layernorm -> normalized/projected to ws + scatter memory/ts outputs.
// 128 threads = 4 waves; each wave owns a 32-wide slice of the N=128 output.
// ---------------------------------------------------------------------------
__global__ __launch_bounds__(128)
void memory_update_kernel(
    const float* __restrict__ memory, const float* __restrict__ memory_ts,
    const float* __restrict__ mail, const float* __restrict__ mail_ts,
    const int* __restrict__ dst_nodes, const float* __restrict__ root_ts,
    const float2* __restrict__ Wt2,
    const float* __restrict__ b_ih, const float* __restrict__ b_hh,
    const float* __restrict__ time_w, const float* __restrict__ time_b,
    const float* __restrict__ ln_gamma, const float* __restrict__ ln_beta,
    const float* __restrict__ tl_w, const float* __restrict__ tl_b,
    float* __restrict__ nrm, float* __restrict__ prj,
    float* __restrict__ out_memory, float* __restrict__ out_memory_ts,
    float* __restrict__ out_mail_ts) {
  __shared__ float Xs[16][K1 + 4];   // pad 4 -> stride 452 (452 % 64 == 4)
  __shared__ float Hs[16][DD + 4];
  __shared__ float mu_s[16], rs_s[16], mts_s[16], tdf_s[16];
  __shared__ int node_s[16];

  const int tid = threadIdx.x;
  const int mrow0 = blockIdx.x * 16;

  // ---- gather phase: 8 threads per row ----
  {
    const int row = tid >> 3;   // 0..15
    const int sub = tid & 7;    // 0..7
    const int grow = mrow0 + row;
    const int node = dst_nodes[grow];
    const float mt = mail_ts[node];
    if (sub == 0) {
      node_s[row] = node;
      mts_s[row] = mt;
      const float t = root_ts[grow & (BB - 1)];
      tdf_s[row] = (t - mt) / (t + 1.0f);
    }
    // mail row: 256 floats -> cols [0,256)
    const float4* mrow = (const float4*)(mail + (long long)node * 256);
    for (int i = 0; i < 8; ++i) {
      float4 v = mrow[sub + i * 8];
      int c = (sub + i * 8) * 4;
      Xs[row][c] = v.x; Xs[row][c + 1] = v.y;
      Xs[row][c + 2] = v.z; Xs[row][c + 3] = v.w;
    }
    // time feature: cos(dt*w + b) -> cols [256,320)
    const float dt = mt - memory_ts[node];
    for (int i = 0; i < 8; ++i) {
      int c = sub * 8 + i;
      Xs[row][256 + c] = cosf(dt * time_w[c] + time_b[c]);
    }
    // prev memory row: 128 floats -> cols [320,448)
    const float4* prow = (const float4*)(memory + (long long)node * DD);
    for (int i = 0; i < 4; ++i) {
      float4 v = prow[sub + i * 8];
      int c = 320 + (sub + i * 8) * 4;
      Xs[row][c] = v.x; Xs[row][c + 1] = v.y;
      Xs[row][c + 2] = v.z; Xs[row][c + 3] = v.w;
    }
  }
  __syncthreads();

  // ---- WMMA GEMM: H(16x128) = X(16x448) @ Wt(448x128) ----
  const int wave = tid >> 5;
  const int lane = tid & 31;
  const int half = lane >> 4;  // 0: K0/K1 , 1: K2/K3 (ISA 16x4 f32 layout)
  const int lid  = lane & 15;  // A: row M ; B: col N
  const int kof  = half * 2;
  const int n0   = wave * 32;

  v8f acc0 = {0.f, 0.f, 0.f, 0.f, 0.f, 0.f, 0.f, 0.f};
  v8f acc1 = acc0;
  for (int k = 0; k < K1; k += 4) {
    v2f a; a.x = Xs[lid][k + kof]; a.y = Xs[lid][k + kof + 1];
    const float2* w = Wt2 + ((k + kof) >> 1) * DD;   // pair index, exact
    float2 t0 = w[n0 + lid];
    float2 t1 = w[n0 + 16 + lid];
    v2f b0; b0.x = t0.x; b0.y = t0.y;
    v2f b1; b1.x = t1.x; b1.y = t1.y;
    acc0 = wmma4(a, b0, acc0);
    acc1 = wmma4(a, b1, acc1);
  }
  // bias + tanh -> LDS (C/D layout: vgpr j, lanes<16 -> M=j, lanes>=16 -> M=j+8)
  for (int j = 0; j < 8; ++j) {
    int row = j + half * 8;
    int c0 = n0 + lid, c1 = n0 + 16 + lid;
    Hs[row][c0] = tanhf(acc0[j] + b_ih[c0] + b_hh[c0]);
    Hs[row][c1] = tanhf(acc1[j] + b_ih[c1] + b_hh[c1]);
  }
  __syncthreads();

  // ---- layernorm stats (one thread per row) ----
  if (tid < 16) {
    float s = 0.f, ss = 0.f;
    for (int c = 0; c < DD; ++c) { float v = Hs[tid][c]; s += v; ss += v * v; }
    float mu = s * (1.0f / DD);
    float var = ss * (1.0f / DD) - mu * mu;
    mu_s[tid] = mu;
    rs_s[tid] = rsqrtf(var + LN_EPS);
  }
  __syncthreads();

  // ---- normalize + project + scatter (tid = column 0..127) ----
  const float g = ln_gamma[tid], be = ln_beta[tid];
  const float tw = tl_w[tid],   tb = tl_b[tid];
  for (int rr = 0; rr < 16; ++rr) {
    const int grow = mrow0 + rr;
    const float nv = g * (Hs[rr][tid] - mu_s[rr]) * rs_s[rr] + be;
    nrm[(long long)grow * DD + tid] = nv;
    prj[(long long)grow * DD + tid] = nv * (1.0f + tdf_s[rr] * tw + tb);
    if (grow < 2 * BB) {
      const int node = node_s[rr];
      out_memory[(long long)node * DD + tid] = nv;
      if (tid == 0) {
        out_memory_ts[node] = mts_s[rr];
        out_mail_ts[node] = root_ts[grow & (BB - 1)];
      }
    }
  }
}

// ---------------------------------------------------------------------------
// new_mail scatter: row r (< 2B) takes normalized[partner] || edge_feat[r % B]
// ---------------------------------------------------------------------------
__global__ __launch_bounds__(256)
void mail_scatter_kernel(const int* __restrict__ dst_nodes,
                         const float* __restrict__ nrm,
                         const float* __restrict__ root_edge_feat,
                         float* __restrict__ out_mail) {
  const int r = blockIdx.x;    // 0 .. 2B-1
  const int c = threadIdx.x;   // 0 .. 255
  const int node = dst_nodes[r];
  const int partner = (r < BB) ? (r + BB) : (r - BB);
  const float v = (c < DD)
      ? nrm[(long long)partner * DD + c]
      : root_edge_feat[(r & (BB - 1)) * DEE + (c - DD)];
  out_mail[(long long)node * 256 + c] = v;
}

// ---------------------------------------------------------------------------
// Link predictor: for 16 edges, pos = relu([src|dst] @ Wp) , neg with neg_h.
// Shared src-half of A; one B fragment feeds both pos & neg accumulators.
// ---------------------------------------------------------------------------
__global__ __launch_bounds__(128)
void predict_kernel(const float* __restrict__ prj,
                    const float2* __restrict__ Wp2,
                    const float* __restrict__ src_b,
                    const float* __restrict__ dst_b,
                    const float* __restrict__ out_w,
                    const float* __restrict__ out_b,
                    float* __restrict__ pos_scores,
                    float* __restrict__ neg_scores) {
  __shared__ float As[16][384 + 4];  // [src(128) | dst(128) | neg(128)]
  __shared__ float Hp[16][DD + 4];
  __shared__ float Hn[16][DD + 4];

  const int tid = threadIdx.x;
  const int r0 = blockIdx.x * 16;

  for (int idx = tid; idx < 16 * 384; idx += 128) {
    const int row = idx / 384, c = idx % 384;
    const int r = r0 + row;
    float v;
    if (c < 128)      v = prj[(long long)r * DD + c];
    else if (c < 256) v = prj[(long long)(r + BB) * DD + (c - 128)];
    else              v = prj[(long long)(r + 2 * BB) * DD + (c - 256)];
    As[row][c] = v;
  }
  __syncthreads();

  const int wave = tid >> 5, lane = tid & 31;
  const int half = lane >> 4, lid = lane & 15;
  const int kof = half * 2;
  const int n0 = wave * 32;

  v8f p0 = {0.f, 0.f, 0.f, 0.f, 0.f, 0.f, 0.f, 0.f};
  v8f p1 = p0, q0 = p0, q1 = p0;
  for (int k = 0; k < K2; k += 4) {
    const int e0 = k + kof, e1 = e0 + 1;
    v2f ap; ap.x = As[lid][e0]; ap.y = As[lid][e1];
    // neg A: second 128 of K comes from the neg slice (cols 256..383)
    const int c0 = (e0 < 128) ? e0 : e0 + 128;
    const int c1 = (e1 < 128) ? e1 : e1 + 128;
    v2f an; an.x = As[lid][c0]; an.y = As[lid][c1];
    const float2* w = Wp2 + (e0 >> 1) * DD;          // pair index, exact
    float2 t0 = w[n0 + lid];
    float2 t1 = w[n0 + 16 + lid];
    v2f b0; b0.x = t0.x; b0.y = t0.y;
    v2f b1; b1.x = t1.x; b1.y = t1.y;
    p0 = wmma4(ap, b0, p0);
    p1 = wmma4(ap, b1, p1);
    q0 = wmma4(an, b0, q0);
    q1 = wmma4(an, b1, q1);
  }
  for (int j = 0; j < 8; ++j) {
    const int row = j + half * 8;
    const int c0 = n0 + lid, c1 = n0 + 16 + lid;
    Hp[row][c0] = fmaxf(p0[j] + src_b[c0] + dst_b[c0], 0.f);
    Hp[row][c1] = fmaxf(p1[j] + src_b[c1] + dst_b[c1], 0.f);
    Hn[row][c0] = fmaxf(q0[j] + src_b[c0] + dst_b[c0], 0.f);
    Hn[row][c1] = fmaxf(q1[j] + src_b[c1] + dst_b[c1], 0.f);
  }
  __syncthreads();

  if (tid < 32) {
    const int row = tid & 15;
    const float* H = (tid < 16) ? Hp[row] : Hn[row];
    float s = 0.f;
    for (int c = 0; c < DD; ++c) s += H[c] * out_w[c];
    s += out_b[0];
    if (tid < 16) pos_scores[r0 + row] = s;
    else          neg_scores[r0 + row] = s;
  }
}

// ---------------------------------------------------------------------------
extern "C" void kernel_launch(void* const* d_in, const int* in_sizes, int n_in,
                              void* d_out, int out_size, void* d_ws,
                              size_t ws_size, hipStream_t stream) {
  const float* memory    = (const float*)d_in[0];
  const float* memory_ts = (const float*)d_in[1];
  const float* mail      = (const float*)d_in[2];
  const float* mail_ts   = (const float*)d_in[3];
  const int*   dst_nodes = (const int*)d_in[4];
  const float* root_ts   = (const float*)d_in[5];
  const float* root_edge_feat = (const float*)d_in[6];
  const float* W_ih   = (const float*)d_in[7];
  const float* W_hh   = (const float*)d_in[8];
  const float* b_ih   = (const float*)d_in[9];
  const float* b_hh   = (const float*)d_in[10];
  const float* time_w = (const float*)d_in[11];
  const float* time_b = (const float*)d_in[12];
  const float* ln_gamma = (const float*)d_in[13];
  const float* ln_beta  = (const float*)d_in[14];
  const float* tl_w   = (const float*)d_in[15];
  const float* tl_b   = (const float*)d_in[16];
  const float* src_w  = (const float*)d_in[17];
  const float* src_b  = (const float*)d_in[18];
  const float* dst_w  = (const float*)d_in[19];
  const float* dst_b  = (const float*)d_in[20];
  const float* out_w  = (const float*)d_in[21];
  const float* out_b  = (const float*)d_in[22];

  // d_out flat layout in reference return order
  float* out = (float*)d_out;
  float* pos_scores    = out;                                   // B
  float* neg_scores    = pos_scores + BB;                       // B
  float* out_memory    = neg_scores + BB;                       // N*128
  float* out_memory_ts = out_memory + (long long)NN * DD;       // N
  float* out_mail      = out_memory_ts + NN;                    // N*256
  float* out_mail_ts   = out_mail + (long long)NN * 256;        // N

  // workspace layout
  float* ws  = (float*)d_ws;
  float* nrm = ws;                              // 3B*128
  float* prj = nrm + (long long)3 * BB * DD;    // 3B*128
  float2* Wt2 = (float2*)(prj + (long long)3 * BB * DD);  // (448/2)*128 pairs
  float2* Wp2 = Wt2 + (K1 / 2) * DD;                      // (256/2)*128 pairs

  // 1) Base copies of the big state arrays (dominant cost: ~3.1 GB HBM traffic)
  hipMemcpyAsync(out_memory, memory, (size_t)NN * DD * sizeof(float),
                 hipMemcpyDeviceToDevice, stream);
  hipMemcpyAsync(out_memory_ts, memory_ts, (size_t)NN * sizeof(float),
                 hipMemcpyDeviceToDevice, stream);
  hipMemcpyAsync(out_mail, mail, (size_t)NN * 256 * sizeof(float),
                 hipMemcpyDeviceToDevice, stream);
  hipMemcpyAsync(out_mail_ts, mail_ts, (size_t)NN * sizeof(float),
                 hipMemcpyDeviceToDevice, stream);

  // 2) Weight transpose + k-pairing for single-b64 WMMA B fragments
  const int prep_elems = (K1 / 2) * DD + (K2 / 2) * DD;
  prep_weights<<<(prep_elems + 255) / 256, 256, 0, stream>>>(
      W_ih, W_hh, src_w, dst_w, Wt2, Wp2);

  // 3) Fused gather + RNN cell (WMMA f32) + layernorm + memory/ts scatter
  memory_update_kernel<<<3 * BB / 16, 128, 0, stream>>>(
      memory, memory_ts, mail, mail_ts, dst_nodes, root_ts, Wt2, b_ih, b_hh,
      time_w, time_b, ln_gamma, ln_beta, tl_w, tl_b, nrm, prj, out_memory,
      out_memory_ts, out_mail_ts);

  // 4) Mail scatter (cross-row dependency on normalized)
  mail_scatter_kernel<<<2 * BB, 256, 0, stream>>>(dst_nodes, nrm,
                                                  root_edge_feat, out_mail);

  // 5) Link predictor (WMMA f32) + score reduction
  predict_kernel<<<BB / 16, 128, 0, stream>>>(prj, Wp2, src_b, dst_b, out_w,
                                              out_b, pos_scores, neg_scores);
}